// decoder_v2_10462540333221
// MI455X (gfx1250) — compile-verified
//
#include <hip/hip_runtime.h>
#include <math.h>

typedef __attribute__((ext_vector_type(2))) float v2f;
typedef __attribute__((ext_vector_type(8))) float v8f;

#define QQ     10
#define CC     64
#define HWSZ   65536      // H*W = 256*256
#define KK     9
#define LL     64800      // 180*360
#define TILE   16
#define PITCH  65         // 64 + 1 pad -> conflict-free [p][c] access

__global__ __launch_bounds__(256)
void decoder_v2_tile_kernel(const float*     __restrict__ x,
                            const long long* __restrict__ idx,
                            const float*     __restrict__ att_w,
                            const float*     __restrict__ att_b,
                            const float*     __restrict__ tc_w,
                            const float*     __restrict__ tc_b,
                            float*           __restrict__ out)
{
    __shared__ float xe[TILE * KK * PITCH];   // gathered tile, [p=l*9+k][c], pitch 65
    __shared__ float featbuf[TILE * 20];      // A matrix: 16 rows x 20 feats (18 real + 2 pad)
    __shared__ float wbuf[20 * 16];           // B matrix: 20 x 16 (zero padded)
    __shared__ float att_lds[TILE * KK];      // sigmoid attention per (l,k)
    __shared__ float tcw_lds[CC * KK];
    __shared__ float tcb_lds[CC];

    const int t    = threadIdx.x;
    const int lane = t & 31;
    const int wv   = t >> 5;
    const int bb   = blockIdx.y;
    const int l0   = blockIdx.x * TILE;

    // ---- stage small weights into LDS, zero WMMA pads ----
    for (int i = t; i < 20 * 16; i += 256) {
        int f = i >> 4, n = i & 15;
        wbuf[i] = (n < KK && f < 18) ? att_w[n * 18 + f] : 0.0f;  // att_w[(c'=n), i, k] flat = n*18+f
    }
    for (int i = t; i < CC * KK; i += 256) tcw_lds[i] = tc_w[i];
    if (t < CC) tcb_lds[t] = tc_b[t];
    if (t < TILE * 2) featbuf[(t >> 1) * 20 + 18 + (t & 1)] = 0.0f;  // zero feature pad 18,19

    // ---- phase 1: random gather + per-(l,k) mean/max over C via wave32 shuffles ----
    const float* xb = x + (size_t)bb * QQ * CC * HWSZ;
    #pragma unroll 2
    for (int it = 0; it < (TILE * KK) / 8; ++it) {       // 18 iters, 8 waves cover 144 (l,k) pairs
        int p = it * 8 + wv;
        int l = p / KK, k = p - l * KK;
        int n = (int)idx[(size_t)(l0 + l) * KK + k];     // idx is int64
        int q = n >> 16, hw = n & (HWSZ - 1);
        const float* col = xb + (size_t)q * CC * HWSZ + hw;
        float v0 = col[(size_t)lane * HWSZ];             // channel = lane
        float v1 = col[(size_t)(lane + 32) * HWSZ];      // channel = lane+32
        xe[p * PITCH + lane]      = v0;
        xe[p * PITCH + lane + 32] = v1;
        float s  = v0 + v1;
        float mx = fmaxf(v0, v1);
        #pragma unroll
        for (int off = 16; off; off >>= 1) {
            s  += __shfl_xor(s, off, 32);
            mx  = fmaxf(mx, __shfl_xor(mx, off, 32));
        }
        if (lane == 0) {
            featbuf[l * 20 + k]      = s * (1.0f / 64.0f);  // mean over C
            featbuf[l * 20 + KK + k] = mx;                  // max over C
        }
    }
    __syncthreads();

    // ---- phase 2: attention logits = stats(16x18) x att_w^T(18x9) via 5x WMMA f32 16x16x4 ----
    if (wv == 0) {
        const int m  = lane & 15;       // A row / D col
        const int kh = lane >> 4;       // K-half select (K=0,1 vs K=2,3)
        v8f acc = {};
        #pragma unroll
        for (int s5 = 0; s5 < 5; ++s5) {                 // K = 20 total (18 real)
            int k0 = s5 * 4 + kh * 2;
            v2f a, b;
            a.x = featbuf[m * 20 + k0];
            a.y = featbuf[m * 20 + k0 + 1];
            b.x = wbuf[k0 * 16 + m];                     // B[k][n], n = lane&15 = m
            b.y = wbuf[(k0 + 1) * 16 + m];
            acc = __builtin_amdgcn_wmma_f32_16x16x4_f32(
                false, a, false, b, (short)0, acc, false, false);
        }
        if (m < KK) {                                    // D col n=m is the output channel c'
            float bv = att_b[m];
            #pragma unroll
            for (int j = 0; j < 8; ++j) {
                int row = j + kh * 8;                    // D row = l within tile
                float lg = acc[j] + bv;
                att_lds[row * KK + m] = 1.0f / (1.0f + expf(-lg));
            }
        }
    }
    __syncthreads();

    // ---- phase 3: out[c, l0+l] = sum_k xe[c,k]*(1+att[l,k])*tc_w[c,k] + tc_b[c] ----
    const int l  = t & 15;       // l fastest across lanes -> coalesced stores
    const int cg = t >> 4;
    #pragma unroll
    for (int j = 0; j < 4; ++j) {
        int c = cg + 16 * j;
        float acc = tcb_lds[c];
        #pragma unroll
        for (int k = 0; k < KK; ++k) {
            float a  = att_lds[l * KK + k];
            float xv = xe[(l * KK + k) * PITCH + c];
            acc = fmaf(xv * (1.0f + a), tcw_lds[c * KK + k], acc);
        }
        out[((size_t)bb * CC + c) * LL + l0 + l] = acc;
    }
}

extern "C" void kernel_launch(void* const* d_in, const int* in_sizes, int n_in,
                              void* d_out, int out_size, void* d_ws, size_t ws_size,
                              hipStream_t stream) {
    (void)in_sizes; (void)n_in; (void)out_size; (void)d_ws; (void)ws_size;
    const float*     x     = (const float*)d_in[0];
    const long long* idx   = (const long long*)d_in[1];   // int64 indices
    const float*     att_w = (const float*)d_in[2];
    const float*     att_b = (const float*)d_in[3];
    const float*     tc_w  = (const float*)d_in[4];
    const float*     tc_b  = (const float*)d_in[5];
    float*           out   = (float*)d_out;

    // grid.x = tiles (fast dim) so batch 0 finishes before batch 1 -> x-slice stays L2 resident
    dim3 grid(LL / TILE, 2);
    decoder_v2_tile_kernel<<<grid, 256, 0, stream>>>(x, idx, att_w, att_b, tc_w, tc_b, out);
}